// WorkingYOLOLoss_57964878627474
// MI455X (gfx1250) — compile-verified
//
#include <hip/hip_runtime.h>
#include <hip/hip_bf16.h>

typedef float v2f __attribute__((ext_vector_type(2)));
typedef float v8f __attribute__((ext_vector_type(8)));

#define BB   64
#define CC   8
#define HH   256
#define WW   256
#define NN   200
#define HW   (HH * WW)          // 65536
#define NTGT (BB * NN)          // 12800
#define BLK  256
#define NBLK (NTGT / BLK)       // 50
#define BBOX_WT 0.05f
#define OBJ_WT  1.0f
#define CLS_WT  0.5f

// Stable softplus == jnp.logaddexp(x, 0)
__device__ __forceinline__ float softplus_f(float x) {
    return fmaxf(x, 0.0f) + log1pf(expf(-fabsf(x)));
}

// Full wave32 sum using V_WMMA_F32_16X16X4_F32.
// A (16x4 f32): lanes 0-15 give rows 0-15 K=0,1 (VGPR0,VGPR1);
//               lanes 16-31 give rows 0-15 K=2,3.
// With a = {v, 0} and B = ones: D[m][n] = v_m + v_{m+16} for all n.
// C/D (16x16 f32): VGPR r holds row r (lanes 0-15) and row r+8 (lanes 16-31).
// Sum of the 8 D VGPRs per lane + xor-16 combine = sum over all 32 lanes.
__device__ __forceinline__ float wave_sum32(float v) {
    v2f a; a[0] = v;    a[1] = 0.0f;
    v2f b; b[0] = 1.0f; b[1] = 1.0f;
    v8f c = {0.0f, 0.0f, 0.0f, 0.0f, 0.0f, 0.0f, 0.0f, 0.0f};
    c = __builtin_amdgcn_wmma_f32_16x16x4_f32(
            /*neg_a=*/false, a, /*neg_b=*/false, b,
            /*c_mod=*/(short)0, c, /*reuse_a=*/false, /*reuse_b=*/false);
    float s = c[0] + c[1] + c[2] + c[3] + c[4] + c[5] + c[6] + c[7];
    s += __shfl_xor(s, 16, 32);
    return s;   // all lanes hold the wave-wide sum
}

__global__ void __launch_bounds__(BLK)
yolo_loss_gather(const float* __restrict__ pred,
                 const float* __restrict__ tgt,
                 float* __restrict__ ws) {
    const int t = blockIdx.x * BLK + threadIdx.x;   // exact: NBLK*BLK == NTGT
    const int bidx = t / NN;

    const float* tp = tgt + (size_t)t * 5;
    const float cls_id = tp[0];
    const float cx = tp[1], cy = tp[2], cw = tp[3], ch = tp[4];

    const bool valid = (cls_id >= 0.0f) && ((cx + cy + cw + ch) > 0.0f);
    const int xpix = (int)(cx * (float)WW);         // trunc toward 0 == astype(int32)
    const int ypix = (int)(cy * (float)HH);
    const bool inb = (xpix >= 0) && (xpix < WW) && (ypix >= 0) && (ypix < HH);

    float bbox = 0.0f, obj = 0.0f, clsl = 0.0f;
    if (valid && inb) {
        const int xs = min(max(xpix, 0), WW - 1);
        const int ys = min(max(ypix, 0), HH - 1);
        const size_t base = (size_t)bidx * (CC * HW) + (size_t)ys * WW + (size_t)xs;
        float p0 = pred[base + 0 * HW];
        float p1 = pred[base + 1 * HW];
        float p2 = pred[base + 2 * HW];
        float p3 = pred[base + 3 * HW];
        float p4 = pred[base + 4 * HW];
        float p5 = pred[base + 5 * HW];
        float p6 = pred[base + 6 * HW];
        float p7 = pred[base + 7 * HW];

        const float d0 = p0 - cx, d1 = p1 - cy, d2 = p2 - cw, d3 = p3 - ch;
        bbox = 0.25f * (d0 * d0 + d1 * d1 + d2 * d2 + d3 * d3);
        obj  = softplus_f(p4) - p4;                 // z = 1.0

        const int cid = (int)fmaxf(cls_id, 0.0f);   // clip(cls_id,0,None)
        const float l0 = softplus_f(p5) - ((cid == 0) ? p5 : 0.0f);
        const float l1 = softplus_f(p6) - ((cid == 1) ? p6 : 0.0f);
        const float l2 = softplus_f(p7) - ((cid == 2) ? p7 : 0.0f);
        clsl = (l0 + l1 + l2) * (1.0f / 3.0f);
    }

    // Reconverged here: EXEC all ones for the WMMA reductions.
    const float wb = wave_sum32(bbox);
    const float wo = wave_sum32(obj);
    const float wc = wave_sum32(clsl);

    __shared__ float part[BLK / 32][3];
    const int lane = threadIdx.x & 31;
    const int wv   = threadIdx.x >> 5;
    if (lane == 0) { part[wv][0] = wb; part[wv][1] = wo; part[wv][2] = wc; }
    __syncthreads();
    if (threadIdx.x == 0) {
        float s0 = 0.0f, s1 = 0.0f, s2 = 0.0f;
        for (int w = 0; w < BLK / 32; ++w) {
            s0 += part[w][0]; s1 += part[w][1]; s2 += part[w][2];
        }
        ws[blockIdx.x * 3 + 0] = s0;
        ws[blockIdx.x * 3 + 1] = s1;
        ws[blockIdx.x * 3 + 2] = s2;
    }
}

__global__ void __launch_bounds__(32)
yolo_loss_finish(const float* __restrict__ ws, float* __restrict__ out) {
    const int lane = threadIdx.x;   // one wave32
    float a0 = 0.0f, a1 = 0.0f, a2 = 0.0f;
    for (int i = lane; i < NBLK; i += 32) {       // fixed order -> deterministic
        a0 += ws[i * 3 + 0];
        a1 += ws[i * 3 + 1];
        a2 += ws[i * 3 + 2];
    }
    // Reconverged: EXEC all ones.
    const float sb = wave_sum32(a0);
    const float so = wave_sum32(a1);
    const float sc = wave_sum32(a2);
    if (lane == 0) {
        out[0] = BBOX_WT * sb + OBJ_WT * so + CLS_WT * sc;
        out[1] = sb;
        out[2] = so;
        out[3] = sc;
    }
}

extern "C" void kernel_launch(void* const* d_in, const int* in_sizes, int n_in,
                              void* d_out, int out_size, void* d_ws, size_t ws_size,
                              hipStream_t stream) {
    const float* pred = (const float*)d_in[0];   // (64, 8, 256, 256) f32
    const float* tgt  = (const float*)d_in[1];   // (64, 200, 5) f32
    float* out = (float*)d_out;                  // 4 f32
    float* ws  = (float*)d_ws;                   // NBLK*3 f32 partials

    yolo_loss_gather<<<NBLK, BLK, 0, stream>>>(pred, tgt, ws);
    yolo_loss_finish<<<1, 32, 0, stream>>>(ws, out);
}